// Three_HMM_62886911148290
// MI455X (gfx1250) — compile-verified
//
#include <hip/hip_runtime.h>

// Fused 3-model neural-HMM forward pass for MI455X (gfx1250, wave32).
// One wave = 16 batch rows; one block = 4 waves (64 rows) + shared weights.
// All GEMMs via V_WMMA_F32_16X16X4_F32. The recursion's logsumexp is rewritten as
// max + exp + f32 GEMM(T_norm) + log (exact).
//
// v5: pin only W1^T (56 VGPRs) and T_norm^T (128 VGPRs) B-operands resident;
// W2^T streams from LDS (software-pipelined by the scheduler). v4 showed that
// pinning all 408 B VGPRs exceeds the ~500-VGPR allocation ceiling and causes
// scratch spills.

#define B_   4096
#define T_   200
#define F_   16
#define H_   64
#define HID_ 100
#define HIDP 112   // HID padded to 7*16 (pads are exact zeros end-to-end)
#define M_   3

typedef __attribute__((ext_vector_type(2))) float v2f;
typedef __attribute__((ext_vector_type(8))) float v8f;

__device__ __forceinline__ v8f wmma_f32(v2f a, v2f b, v8f c) {
  // D(16x16,f32) = A(16x4,f32) * B(4x16,f32) + C
  return __builtin_amdgcn_wmma_f32_16x16x4_f32(
      /*neg_a=*/false, a, /*neg_b=*/false, b,
      /*c_mod=*/(short)0, c, /*reuse_a=*/false, /*reuse_b=*/false);
}

// Opacity pin: forces the value to stay register-resident (no remat from LDS).
__device__ __forceinline__ void pin(v2f& v) { asm("" : "+v"(v)); }

// Reductions over the 16 lanes of each wave-half (masks < 16 never cross halves).
__device__ __forceinline__ float hmax16(float v) {
  v = fmaxf(v, __shfl_xor(v, 1));
  v = fmaxf(v, __shfl_xor(v, 2));
  v = fmaxf(v, __shfl_xor(v, 4));
  v = fmaxf(v, __shfl_xor(v, 8));
  return v;
}
__device__ __forceinline__ float hsum16(float v) {
  v += __shfl_xor(v, 1);
  v += __shfl_xor(v, 2);
  v += __shfl_xor(v, 4);
  v += __shfl_xor(v, 8);
  return v;
}

__device__ __forceinline__ v8f splat8(float x) {
  v8f r;
#pragma unroll
  for (int i = 0; i < 8; ++i) r[i] = x;
  return r;
}

__launch_bounds__(128, 1)
__global__ void nhmm3_kernel(const float* __restrict__ x,
                             const float* __restrict__ W1,
                             const float* __restrict__ b1,
                             const float* __restrict__ W2,
                             const float* __restrict__ b2,
                             const float* __restrict__ trans,
                             const float* __restrict__ priors,
                             float* __restrict__ out) {
  // Pair layout: s-th K-slice, half h holds {B[4s+2h][col], B[4s+2h+1][col]}.
  __shared__ v2f   sW1P[4][2][112];     // GEMM1 B staging
  __shared__ v2f   sW2P[28][2][80];     // GEMM2 B (streamed every iteration)
  __shared__ v2f   sTTP[16][2][80];     // GEMM3 B staging
  __shared__ float sB1[HIDP];
  __shared__ float sB2[H_];
  __shared__ float sLP[H_];             // log_softmax(priors)
  __shared__ float sCM[H_], sCI[H_];    // per-column max / inv-sum of trans
  __shared__ float sStageH[4][16][114]; // per-wave h bounce (emission)
  __shared__ float sStageP[4][16][66];  // per-wave p bounce (recursion)

  const int tid = threadIdx.x;
  const int m   = blockIdx.y;

  // ---------------- one-time cooperative weight staging ----------------
  const float* W1m = W1 + (size_t)m * HID_ * F_;
  for (int i = tid; i < 4 * 2 * HIDP; i += 128) {
    const int n = i % HIDP, sh = i / HIDP, h = sh & 1, s = sh >> 1;
    const int k0 = 4 * s + 2 * h;
    v2f v;
    v.x = (n < HID_) ? W1m[n * F_ + k0]     : 0.f;
    v.y = (n < HID_) ? W1m[n * F_ + k0 + 1] : 0.f;
    sW1P[s][h][n] = v;
  }
  for (int n = tid; n < HIDP; n += 128) sB1[n] = (n < HID_) ? b1[m * HID_ + n] : 0.f;

  const float* W2m = W2 + (size_t)m * H_ * HID_;
  for (int i = tid; i < 28 * 2 * H_; i += 128) {
    const int c = i % H_, sh = i / H_, h = sh & 1, s = sh >> 1;
    const int k0 = 4 * s + 2 * h;
    v2f v;
    v.x = (k0     < HID_) ? W2m[c * HID_ + k0]     : 0.f;
    v.y = (k0 + 1 < HID_) ? W2m[c * HID_ + k0 + 1] : 0.f;
    sW2P[s][h][c] = v;
  }
  for (int s = tid; s < H_; s += 128) sB2[s] = b2[m * H_ + s];

  const float* trm = trans + (size_t)m * H_ * H_;
  if (tid < H_) {
    const int j = tid;                      // column of trans (softmax over axis 0)
    float cm = -__builtin_inff();
    for (int i = 0; i < H_; ++i) cm = fmaxf(cm, trm[i * H_ + j]);
    float s = 0.f;
    for (int i = 0; i < H_; ++i) s += __expf(trm[i * H_ + j] - cm);
    sCM[j] = cm;
    sCI[j] = 1.f / s;

    const float* pr = priors + m * H_;      // redundant tiny per-thread logsumexp
    float pm = -__builtin_inff();
    for (int i = 0; i < H_; ++i) pm = fmaxf(pm, pr[i]);
    float ps = 0.f;
    for (int i = 0; i < H_; ++i) ps += __expf(pr[i] - pm);
    sLP[j] = pr[j] - pm - __logf(ps);
  }
  __syncthreads();
  // T_norm^T pairs: B element (K=j, N=i) = softmax_col(trans)[i][j]
  for (int idx = tid; idx < 16 * 2 * H_; idx += 128) {
    const int c = idx % H_, sh = idx / H_, h = sh & 1, s = sh >> 1;
    const int j0 = 4 * s + 2 * h;
    v2f v;
    v.x = __expf(trm[c * H_ + j0]     - sCM[j0])     * sCI[j0];
    v.y = __expf(trm[c * H_ + j0 + 1] - sCM[j0 + 1]) * sCI[j0 + 1];
    sTTP[s][h][c] = v;
  }
  __syncthreads();

  // ---------------- per-wave setup ----------------
  const int lane = tid & 31;
  const int wv   = tid >> 5;
  const int hf   = lane >> 4;     // wave half: selects the {K+2,K+3} plane
  const int l16  = lane & 15;
  const long bBase = (long)blockIdx.x * 64 + wv * 16;      // first batch row of tile
  const float* xrow = x + (size_t)(bBase + l16) * T_ * F_; // this lane's A-row of x

  float (*stgH)[114] = sStageH[wv];
  float (*stgP)[66]  = sStageP[wv];

  // ---- pin the small/critical loop-invariant B operands in VGPRs ----
  v2f rW1[4][7];     //  56 VGPRs (emission front-end)
  v2f rTT[16][4];    // 128 VGPRs (serial recursion critical path)
#pragma unroll
  for (int s4 = 0; s4 < 4; ++s4)
#pragma unroll
    for (int nt = 0; nt < 7; ++nt) {
      rW1[s4][nt] = sW1P[s4][hf][nt * 16 + l16];
      pin(rW1[s4][nt]);
    }
#pragma unroll
  for (int s4 = 0; s4 < 16; ++s4)
#pragma unroll
    for (int st = 0; st < 4; ++st) {
      rTT[s4][st] = sTTP[s4][hf][st * 16 + l16];
      pin(rTT[s4][st]);
    }

  const float rb2[4] = { sB2[l16], sB2[16 + l16], sB2[32 + l16], sB2[48 + l16] };
  const float rb1[7] = { sB1[l16],      sB1[16 + l16], sB1[32 + l16], sB1[48 + l16],
                         sB1[64 + l16], sB1[80 + l16], sB1[96 + l16] };

  // Emission for one timestep: em[4] C-tiles of log-softmaxed logits (16 x 64).
  auto emit = [&](int t, v8f em[4]) {
    const float* xt = xrow + (size_t)t * F_;

    // GEMM1: h(16x112) = x(16x16) * W1^T(16x112)
    v8f h[7];
#pragma unroll
    for (int nt = 0; nt < 7; ++nt) h[nt] = splat8(0.f);
#pragma unroll
    for (int s4 = 0; s4 < 4; ++s4) {
      v2f a = *(const v2f*)(xt + 4 * s4 + 2 * hf);   // global b64, A f32 layout
#pragma unroll
      for (int nt = 0; nt < 7; ++nt)
        h[nt] = wmma_f32(a, rW1[s4][nt], h[nt]);
    }
    // bias + ReLU, stage in A-row-major for GEMM2
#pragma unroll
    for (int nt = 0; nt < 7; ++nt) {
#pragma unroll
      for (int v = 0; v < 8; ++v)                    // C-tile row = v + 8*half
        stgH[v + 8 * hf][nt * 16 + l16] = fmaxf(h[nt][v] + rb1[nt], 0.f);
    }

    // GEMM2: logits(16x64) = h(16x112) * W2^T(112x64) + b2  (B streamed from LDS)
#pragma unroll
    for (int st = 0; st < 4; ++st) em[st] = splat8(rb2[st]);
#pragma unroll
    for (int s4 = 0; s4 < 28; ++s4) {
      v2f a = *(const v2f*)&stgH[l16][4 * s4 + 2 * hf];   // aligned ds b64
#pragma unroll
      for (int st = 0; st < 4; ++st)
        em[st] = wmma_f32(a, sW2P[s4][hf][st * 16 + l16], em[st]);
    }

    // log_softmax over 64 states per batch row
#pragma unroll
    for (int v = 0; v < 8; ++v) {
      float mx = fmaxf(fmaxf(em[0][v], em[1][v]), fmaxf(em[2][v], em[3][v]));
      mx = hmax16(mx);
      float s = __expf(em[0][v] - mx) + __expf(em[1][v] - mx) +
                __expf(em[2][v] - mx) + __expf(em[3][v] - mx);
      s = hsum16(s);
      const float lz = mx + __logf(s);
      em[0][v] -= lz; em[1][v] -= lz; em[2][v] -= lz; em[3][v] -= lz;
    }
  };

  v8f alpha[4];   // log-alpha, 16 rows x 64 states as 4 WMMA C-tiles
  v8f em[4];

  // ---- t = 0 (peeled): alpha = em + log_softmax(priors) ----
  emit(0, em);
  {
    const float rlp[4] = { sLP[l16], sLP[16 + l16], sLP[32 + l16], sLP[48 + l16] };
#pragma unroll
    for (int st = 0; st < 4; ++st)
#pragma unroll
      for (int v = 0; v < 8; ++v) alpha[st][v] = em[st][v] + rlp[st];
  }

  // ---- steady-state: emission + recursion ----
#pragma unroll 1
  for (int t = 1; t < T_; ++t) {
    // Speculative prefetch of next step's x row (OOB prefetch silently dropped).
    __builtin_prefetch(xrow + (size_t)(t + 1) * F_, 0, 0);  // global_prefetch_b8

    emit(t, em);

    // alpha' = em + c + log( exp(alpha - c) * T_norm^T ),  c = rowmax(alpha)
    float cmax[8];
#pragma unroll
    for (int v = 0; v < 8; ++v) {
      float mx = fmaxf(fmaxf(alpha[0][v], alpha[1][v]), fmaxf(alpha[2][v], alpha[3][v]));
      cmax[v] = hmax16(mx);
#pragma unroll
      for (int st = 0; st < 4; ++st)
        stgP[v + 8 * hf][st * 16 + l16] = __expf(alpha[st][v] - cmax[v]);
    }
    v8f acc[4];
#pragma unroll
    for (int st = 0; st < 4; ++st) acc[st] = splat8(0.f);
#pragma unroll
    for (int s4 = 0; s4 < 16; ++s4) {
      v2f a = *(const v2f*)&stgP[l16][4 * s4 + 2 * hf];   // aligned ds b64
#pragma unroll
      for (int st = 0; st < 4; ++st)
        acc[st] = wmma_f32(a, rTT[s4][st], acc[st]);
    }
#pragma unroll
    for (int st = 0; st < 4; ++st)
#pragma unroll
      for (int v = 0; v < 8; ++v)
        alpha[st][v] = em[st][v] + cmax[v] + __logf(acc[st][v]);
  }

  // ---- final logsumexp over states -> out[b, m] ----
#pragma unroll
  for (int v = 0; v < 8; ++v) {
    float mx = fmaxf(fmaxf(alpha[0][v], alpha[1][v]), fmaxf(alpha[2][v], alpha[3][v]));
    mx = hmax16(mx);
    float s = __expf(alpha[0][v] - mx) + __expf(alpha[1][v] - mx) +
              __expf(alpha[2][v] - mx) + __expf(alpha[3][v] - mx);
    s = hsum16(s);
    const float res = mx + __logf(s);
    if (l16 == 0) out[(bBase + v + 8 * hf) * 3 + m] = res;
  }
}

extern "C" void kernel_launch(void* const* d_in, const int* in_sizes, int n_in,
                              void* d_out, int out_size, void* d_ws, size_t ws_size,
                              hipStream_t stream) {
  const float* x      = (const float*)d_in[0];
  const float* W1     = (const float*)d_in[1];
  const float* b1     = (const float*)d_in[2];
  const float* W2     = (const float*)d_in[3];
  const float* b2     = (const float*)d_in[4];
  const float* trans  = (const float*)d_in[5];
  const float* priors = (const float*)d_in[6];
  float* out = (float*)d_out;

  dim3 grid(B_ / 64, M_);   // 64 batch-tiles x 3 models = 192 blocks, 4 waves each
  dim3 block(128);
  hipLaunchKernelGGL(nhmm3_kernel, grid, block, 0, stream,
                     x, W1, b1, W2, b2, trans, priors, out);
}